// GCF_76587856822393
// MI455X (gfx1250) — compile-verified
//
#include <hip/hip_runtime.h>

#define B_ 16
#define L_ 32
#define H_ 768
#define E_ 200000
#define R_ 512
#define T_ 2000000
#define EPS_ 1e-6f

typedef __attribute__((ext_vector_type(2))) float v2f;
typedef __attribute__((ext_vector_type(8))) float v8f;

// ---------------------------------------------------------------------------
// WMMA f32 GEMM: per batch, C(L_ x N) = A(L_ x K) @ W(K x N) + bias(N)
// One wave computes one 16x16 tile using V_WMMA_F32_16X16X4_F32 (exact fp32).
// grid = (num_tiles/8, B_), block = 256 (8 waves)
// A-frag (16x4 f32): lanes 0-15 -> M, VGPR0=K0,VGPR1=K1; lanes 16-31 -> K2,K3
// B-frag (4x16 f32): lanes 0-15 -> N, VGPR0=K0,VGPR1=K1; lanes 16-31 -> K2,K3
// C/D: VGPR j: lanes 0-15 -> M=j, N=lane; lanes 16-31 -> M=j+8, N=lane-16
// ---------------------------------------------------------------------------
__global__ void k_gemm_wmma(const float* __restrict__ A, const float* __restrict__ W,
                            const float* __restrict__ bias, float* __restrict__ C,
                            int K, int N, int strideA, int strideC) {
  int b    = blockIdx.y;
  int wave = threadIdx.x >> 5;
  int lane = threadIdx.x & 31;
  int tile = blockIdx.x * 8 + wave;
  int n_tiles = N >> 4;
  int m_t = tile / n_tiles;          // wave-uniform
  int n_t = tile % n_tiles;
  if (m_t >= (L_ >> 4)) return;      // uniform per wave -> EXEC stays all-1s

  const float* Ab = A + (size_t)b * strideA;
  float*       Cb = C + (size_t)b * strideC;
  int row = lane & 15;
  int kh  = (lane >> 4) << 1;        // 0 or 2
  int m0 = m_t << 4, n0 = n_t << 4;

  // A pair (k+kh, k+kh+1) is contiguous & 8B aligned -> load as float2
  const float2* Arow = (const float2*)(Ab + (size_t)(m0 + row) * K);

  v8f acc = {};
  for (int k = 0; k < K; k += 4) {
    float2 ap = Arow[(k + kh) >> 1];
    v2f a, w;
    a[0] = ap.x;
    a[1] = ap.y;
    w[0] = W[(size_t)(k + kh) * N + n0 + row];
    w[1] = W[(size_t)(k + kh + 1) * N + n0 + row];
    acc = __builtin_amdgcn_wmma_f32_16x16x4_f32(false, a, false, w,
                                                (short)0, acc, false, false);
  }
  int rbase = m0 + ((lane >> 4) << 3);
  float bv = bias[n0 + row];
#pragma unroll
  for (int j = 0; j < 8; ++j)
    Cb[(size_t)(rbase + j) * N + n0 + row] = acc[j] + bv;
}

// ---------------------------------------------------------------------------
// Attention: q_logits = qwh . hkey^T per batch, softmax over j, mask+renorm,
// write q_dist and its column sums. grid = B_, block = 1024 (wave per row i).
// ---------------------------------------------------------------------------
__global__ void k_attn(const float* __restrict__ qwh, const float* __restrict__ hkey,
                       const float* __restrict__ mask, float* __restrict__ qdist,
                       float* __restrict__ colsum) {
  int b = blockIdx.x;
  int i = threadIdx.x >> 5;
  int j = threadIdx.x & 31;
  const float4* q4 = (const float4*)(qwh + ((size_t)b * L_ + i) * H_);
  const float4* k4 = (const float4*)(hkey + ((size_t)b * L_ + j) * H_);
  float dot = 0.f;
  for (int h = 0; h < H_ / 4; ++h) {
    float4 a = q4[h], c = k4[h];
    dot += a.x * c.x + a.y * c.y + a.z * c.z + a.w * c.w;
  }
  float m = dot;
  for (int off = 16; off; off >>= 1) m = fmaxf(m, __shfl_xor(m, off, 32));
  float p = __expf(dot - m);
  float s = p;
  for (int off = 16; off; off >>= 1) s += __shfl_xor(s, off, 32);
  float d = (p / s) * mask[b * L_ + j];
  float rs = d;
  for (int off = 16; off; off >>= 1) rs += __shfl_xor(rs, off, 32);
  float qd = d / (rs + EPS_);
  qdist[((size_t)b * L_ + i) * L_ + j] = qd;

  __shared__ float cs[L_];
  if (threadIdx.x < L_) cs[threadIdx.x] = 0.f;
  __syncthreads();
  atomicAdd(&cs[j], qd);
  __syncthreads();
  if (threadIdx.x < L_) colsum[b * L_ + threadIdx.x] = cs[threadIdx.x];
}

// hop_ctx[b,i,h] = sum_j q_dist[b,i,j] * q_hop[b,j,h]
__global__ void k_hopctx(const float* __restrict__ qdist, const float* __restrict__ qhop,
                         float* __restrict__ hopctx) {
  int idx = blockIdx.x * blockDim.x + threadIdx.x;   // B_*L_*H_ exact
  int h  = idx % H_;
  int bi = idx / H_;
  int b  = bi / L_;
  const float* qd = qdist + (size_t)bi * L_;
  const float* qh = qhop + (size_t)b * L_ * H_ + h;
  float acc = 0.f;
#pragma unroll 8
  for (int j = 0; j < L_; ++j) acc += qd[j] * qh[(size_t)j * H_];
  hopctx[idx] = acc;
}

// q_hop / prev_ctx update (elementwise)
__global__ void k_update(const float* __restrict__ qwh, const float* __restrict__ hopctx,
                         const float* __restrict__ zlin, float* __restrict__ qhop,
                         float* __restrict__ prevctx, int step) {
  int idx = blockIdx.x * blockDim.x + threadIdx.x;   // B_*L_*H_ exact
  float hc = hopctx[idx];
  if (step == 0) {
    qhop[idx]    = qwh[idx] + hc;
    prevctx[idx] = hc;
  } else {
    float z  = 1.f / (1.f + __expf(-zlin[idx]));
    float zp = z * prevctx[idx];
    qhop[idx]    = qwh[idx] + hc + zp;
    prevctx[idx] = hc + zp;
  }
}

// word attention (softmax over column sums) + ctx_h.  grid=B_, block=768
__global__ void k_attctx(const float* __restrict__ colsum, const float* __restrict__ mask,
                         const float* __restrict__ qhop, float* __restrict__ ctxhist,
                         int step) {
  int b = blockIdx.x;
  __shared__ float att[L_];
  if (threadIdx.x < 32) {
    int j = threadIdx.x;
    float v = colsum[b * L_ + j];
    float m = v;
    for (int off = 16; off; off >>= 1) m = fmaxf(m, __shfl_xor(m, off, 32));
    float p = __expf(v - m);
    float s = p;
    for (int off = 16; off; off >>= 1) s += __shfl_xor(s, off, 32);
    float d = (p / s) * mask[b * L_ + j];
    float rs = d;
    for (int off = 16; off; off >>= 1) rs += __shfl_xor(rs, off, 32);
    att[j] = d / (rs + EPS_);
  }
  __syncthreads();
  int h = threadIdx.x;
  float acc = 0.f;
#pragma unroll 8
  for (int l = 0; l < L_; ++l) acc += att[l] * qhop[((size_t)b * L_ + l) * H_ + h];
  ctxhist[((size_t)b * 2 + step) * H_ + h] = acc;
}

// rel_dist, written TRANSPOSED: reldT[r*16 + b] = sigmoid(ctx_h[b] . rel_w[:,r] + rel_b[r])
__global__ void k_reldist(const float* __restrict__ ctxhist, const float* __restrict__ relw,
                          const float* __restrict__ relb, float* __restrict__ reldT,
                          int step) {
  int idx = blockIdx.x * blockDim.x + threadIdx.x;   // R_*B_ exact
  int b = idx & (B_ - 1);
  int r = idx >> 4;
  const float* c = ctxhist + ((size_t)b * 2 + step) * H_;
  float acc = relb[r];
  for (int h = 0; h < H_; ++h) acc += c[h] * relw[(size_t)h * R_ + r];
  reldT[idx] = 1.f / (1.f + __expf(-acc));
}

// heads (B,E) -> headsT (E,16): one 64B line per entity
__global__ void k_transposeE(const float* __restrict__ src, float* __restrict__ dstT) {
  int e = blockIdx.x * blockDim.x + threadIdx.x;
  if (e >= E_) return;
  float4* out = (float4*)(dstT + (size_t)e * B_);
#pragma unroll
  for (int q = 0; q < 4; ++q) {
    float4 v;
    v.x = src[(size_t)(q * 4 + 0) * E_ + e];
    v.y = src[(size_t)(q * 4 + 1) * E_ + e];
    v.z = src[(size_t)(q * 4 + 2) * E_ + e];
    v.w = src[(size_t)(q * 4 + 3) * E_ + e];
    out[q] = v;
  }
}

__global__ void k_zero(float* __restrict__ p, int n) {
  int i = blockIdx.x * blockDim.x + threadIdx.x;
  if (i < n) p[i] = 0.f;
}

// The hot kernel: 2M triples x 16 batches of fp32 atomic adds, L2 resident.
// Entity-major [e][16] layout: gather = 4 x float4 from one cacheline,
// rel_dist table = 32 KB, stays hot in WGP$/L0 (read as float4 directly);
// scatter = 16 no-return global_atomic_add_f32 into one cacheline.
__global__ void k_scatter(const int* __restrict__ subj, const int* __restrict__ rel,
                          const int* __restrict__ obj, const float* __restrict__ srcT,
                          const float* __restrict__ reldT, float* __restrict__ dstT) {
  int t = blockIdx.x * blockDim.x + threadIdx.x;
  if (t >= T_) return;
  int s = subj[t], r = rel[t], o = obj[t];
  const float4* sp = (const float4*)(srcT + (size_t)s * B_);
  const float4* rp = (const float4*)(reldT + (size_t)r * B_);
  float* dp = dstT + (size_t)o * B_;
#pragma unroll
  for (int q = 0; q < 4; ++q) {
    float4 sv = sp[q];
    float4 rv = rp[q];
    unsafeAtomicAdd(&dp[q * 4 + 0], sv.x * rv.x);
    unsafeAtomicAdd(&dp[q * 4 + 1], sv.y * rv.y);
    unsafeAtomicAdd(&dp[q * 4 + 2], sv.z * rv.z);
    unsafeAtomicAdd(&dp[q * 4 + 3], sv.w * rv.w);
  }
}

// last_e = new_e / max(new_e, 1)  ==  min(new_e, 1) for non-negative values
__global__ void k_clamp(float* __restrict__ p, int n) {
  int i = blockIdx.x * blockDim.x + threadIdx.x;
  if (i < n) {
    float v = p[i];
    p[i] = v > 1.f ? 1.f : v;
  }
}

// hop attention: softmax over the two hops of ctx_hist @ hop_w + hop_b. 32 threads.
__global__ void k_hopattn(const float* __restrict__ ctxhist, const float* __restrict__ hopw,
                          const float* __restrict__ hopb, float* __restrict__ attn) {
  int tid = threadIdx.x;            // tt*16 + b
  int tt = tid >> 4, b = tid & 15;
  const float* c = ctxhist + ((size_t)b * 2 + tt) * H_;
  float acc = hopb[0];
  for (int h = 0; h < H_; ++h) acc += c[h] * hopw[h];
  float other = __shfl_xor(acc, 16, 32);
  float m = fmaxf(acc, other);
  float e0 = __expf(acc - m);
  float e1 = __expf(other - m);
  attn[b * 2 + tt] = e0 / (e0 + e1);
}

// e_score[b,e] = a0*hop0T[e,b] + a1*min(hop1T[e,b],1)   (clamp of step-1 folded in)
__global__ void k_final(const float* __restrict__ h0T, const float* __restrict__ h1T,
                        const float* __restrict__ attn, float* __restrict__ out) {
  long i = (long)blockIdx.x * blockDim.x + threadIdx.x;
  if (i >= (long)B_ * E_) return;
  int b = (int)(i / E_);
  long e = i % E_;
  float v1 = h1T[e * B_ + b];
  v1 = v1 > 1.f ? 1.f : v1;
  out[i] = attn[b * 2 + 0] * h0T[e * B_ + b] + attn[b * 2 + 1] * v1;
}

extern "C" void kernel_launch(void* const* d_in, const int* in_sizes, int n_in,
                              void* d_out, int out_size, void* d_ws, size_t ws_size,
                              hipStream_t stream) {
  (void)in_sizes; (void)n_in; (void)out_size; (void)ws_size;
  const float* heads  = (const float*)d_in[0];
  const float* qwh    = (const float*)d_in[1];
  const float* mask   = (const float*)d_in[2];
  const int*   subj   = (const int*)d_in[3];
  const int*   rel    = (const int*)d_in[4];
  const int*   obj    = (const int*)d_in[5];
  const float* key_w  = (const float*)d_in[6];
  const float* key_b  = (const float*)d_in[7];
  const float* rel_w  = (const float*)d_in[8];
  const float* rel_b  = (const float*)d_in[9];
  const float* hop_w  = (const float*)d_in[10];
  const float* hop_b  = (const float*)d_in[11];
  const float* hw_w   = (const float*)d_in[12];
  const float* hw_b   = (const float*)d_in[13];
  float* out = (float*)d_out;

  float* ws = (float*)d_ws;
  float* headsT  = ws;                       // E_*B_ = 3,200,000
  float* hopT0   = headsT + (size_t)E_ * B_; // 3,200,000
  float* hopT1   = hopT0 + (size_t)E_ * B_;  // 3,200,000
  float* qhop    = hopT1 + (size_t)E_ * B_;  // 393,216
  float* prevctx = qhop + B_ * L_ * H_;
  float* hkey    = prevctx + B_ * L_ * H_;
  float* hopctx  = hkey + B_ * L_ * H_;
  float* zlin    = hopctx + B_ * L_ * H_;
  float* qdist   = zlin + B_ * L_ * H_;      // 16,384
  float* colsum  = qdist + B_ * L_ * L_;     // 512
  float* ctxhist = colsum + B_ * L_;         // 24,576
  float* reldT   = ctxhist + B_ * 2 * H_;    // 8,192
  float* attn    = reldT + R_ * B_;          // 32

  const int BLH = B_ * L_ * H_;              // 393,216
  const int BE  = B_ * E_;                   // 3,200,000
  dim3 gemmGrid((2 * (H_ / 16)) / 8, B_);    // 96 tiles, 8 waves/block

  k_transposeE<<<(E_ + 255) / 256, 256, 0, stream>>>(heads, headsT);
  k_zero<<<(BE + 255) / 256, 256, 0, stream>>>(hopT0, BE);
  k_zero<<<(BE + 255) / 256, 256, 0, stream>>>(hopT1, BE);

  // ---------------- step 0 ----------------
  k_gemm_wmma<<<gemmGrid, 256, 0, stream>>>(qwh, key_w, key_b, hkey, H_, H_, L_ * H_, L_ * H_);
  k_attn<<<B_, 1024, 0, stream>>>(qwh, hkey, mask, qdist, colsum);
  k_hopctx<<<BLH / 256, 256, 0, stream>>>(qdist, qwh, hopctx);
  k_update<<<BLH / 256, 256, 0, stream>>>(qwh, hopctx, zlin, qhop, prevctx, 0);
  k_attctx<<<B_, H_, 0, stream>>>(colsum, mask, qhop, ctxhist, 0);
  k_reldist<<<(R_ * B_) / 256, 256, 0, stream>>>(ctxhist, rel_w, rel_b, reldT, 0);
  k_scatter<<<(T_ + 255) / 256, 256, 0, stream>>>(subj, rel, obj, headsT, reldT, hopT0);
  k_clamp<<<(BE + 255) / 256, 256, 0, stream>>>(hopT0, BE);

  // ---------------- step 1 ----------------
  k_gemm_wmma<<<gemmGrid, 256, 0, stream>>>(qhop, key_w, key_b, hkey, H_, H_, L_ * H_, L_ * H_);
  k_attn<<<B_, 1024, 0, stream>>>(qwh, hkey, mask, qdist, colsum);
  k_hopctx<<<BLH / 256, 256, 0, stream>>>(qdist, qhop, hopctx);
  k_gemm_wmma<<<gemmGrid, 256, 0, stream>>>(prevctx, hw_w, hw_b, zlin, H_, H_, L_ * H_, L_ * H_);
  k_update<<<BLH / 256, 256, 0, stream>>>(qwh, hopctx, zlin, qhop, prevctx, 1);
  k_attctx<<<B_, H_, 0, stream>>>(colsum, mask, qhop, ctxhist, 1);
  k_reldist<<<(R_ * B_) / 256, 256, 0, stream>>>(ctxhist, rel_w, rel_b, reldT, 1);
  k_scatter<<<(T_ + 255) / 256, 256, 0, stream>>>(subj, rel, obj, hopT0, reldT, hopT1);

  // ---------------- finale (step-1 clamp folded into k_final) ----------------
  k_hopattn<<<1, 32, 0, stream>>>(ctxhist, hop_w, hop_b, attn);
  k_final<<<(BE + 255) / 256, 256, 0, stream>>>(hopT0, hopT1, attn, out);
}